// Modulated_SpectralConv2d_78340203479305
// MI455X (gfx1250) — compile-verified
//
#include <hip/hip_runtime.h>
#include <hip/hip_bf16.h>
#include <math.h>

// ---------------------------------------------------------------------------
// Modulated SpectralConv2d for MI455X (gfx1250), decomposed into WMMA GEMMs.
// - All matrix math: V_WMMA_F32_16X16X4_F32 (wave32 WMMA)
// - Basis matrices stored pre-swizzled in WMMA fragment order (b64 loads)
// - Point-major spectral layouts for contiguous fragment loads in the mix
// - Tensor Data Mover (tensor_load_to_lds) feeds LDS A-tiles where available
// ---------------------------------------------------------------------------

typedef float v2f __attribute__((ext_vector_type(2)));
typedef float v8f __attribute__((ext_vector_type(8)));

__device__ __forceinline__ v8f wmma4(v2f a, v2f b, v8f c) {
  // D = A(16x4) * B(4x16) + C(16x16), all f32
  return __builtin_amdgcn_wmma_f32_16x16x4_f32(false, a, false, b, (short)0, c,
                                               false, false);
}

#define TWO_PI 6.283185307179586476925f

// Problem constants
#define BATCH 8
#define CI 32
#define CO 32

// Workspace layout (float offsets)
#define OFF_ROWB   ((size_t)0)            // [4][64][32][2]  fwd row basis (fragment order)
#define OFF_COLA   ((size_t)16384)        // [64][256][2]    fwd col basis (complex)
#define OFF_ICOLA  ((size_t)49152)        // [256][64][2]    inv col basis (complex)
#define OFF_IROWB  ((size_t)81920)        // [16][16][32][2] inv row basis (fragment order)
#define OFF_WC     ((size_t)131072)       // [2][32][32][32][32][2] point-major spectral wts
#define OFF_XW     (OFF_WC + 4194304)     // [B*Ci*256][64] row-DFT result (re,im)
#define OFF_Z      OFF_XW                 // reuse: [B*Co*256][64] inv-col result
#define OFF_XF     (OFF_XW + 4194304)     // [64*32][8][32][2] point-major truncated spectrum
#define OFF_YF     (OFF_XF + 1048576)     // [B*Co][64][32][2] mixed spectrum

// ---------------------------------------------------------------------------
// TDM support (guarded: falls back to plain LDS staging if builtin absent)
// ---------------------------------------------------------------------------
#if defined(__gfx1250__) && __has_builtin(__builtin_amdgcn_tensor_load_to_lds) && \
    __has_builtin(__builtin_amdgcn_s_wait_tensorcnt)
#define HAVE_TDM 1
#else
#define HAVE_TDM 0
#endif

#if HAVE_TDM
typedef unsigned int u32x4 __attribute__((ext_vector_type(4)));
typedef int i32x4 __attribute__((ext_vector_type(4)));
typedef int i32x8 __attribute__((ext_vector_type(8)));

// Issue a 2D f32 tile DMA: rows x width elements, row stride == width,
// from global gsrc into LDS offset lds_off. Wave-uniform arguments.
__device__ __forceinline__ void tdm_load_2d_f32(const float* gsrc, unsigned lds_off,
                                                unsigned width, unsigned rows) {
  unsigned long long ga = (unsigned long long)(const void*)gsrc;
  u32x4 g0 = {};
  g0[0] = 1u;                                          // count=1, user descriptor
  g0[1] = lds_off;                                     // lds_addr
  g0[2] = (unsigned)(ga & 0xFFFFFFFFu);                // global_addr[31:0]
  g0[3] = (unsigned)((ga >> 32) & 0x01FFFFFFu) | (2u << 30);  // [56:32] + type=2
  i32x8 g1 = {};
  g1[0] = (2 << 16);                                   // data_size = 4 bytes
  g1[1] = (int)(width << 16);                          // tensor_dim0 (lo 16)
  g1[2] = (int)(rows << 16);                           // tensor_dim1 (lo 16)
  g1[3] = (int)(width << 16);                          // tile_dim0
  g1[4] = (int)rows;                                   // tile_dim1
  g1[5] = (int)width;                                  // tensor_dim0_stride (lo 32)
  i32x4 gz = {};
#if __clang_major__ >= 23
  i32x8 gz8 = {};
  __builtin_amdgcn_tensor_load_to_lds(g0, g1, gz, gz, gz8, 0);
#else
  __builtin_amdgcn_tensor_load_to_lds(g0, g1, gz, gz, 0);
#endif
}
#endif

// ---------------------------------------------------------------------------
// Kernel 1: generate DFT basis tables (fragment-swizzled where consumed so)
// ---------------------------------------------------------------------------
__global__ void gen_basis(float* __restrict__ rowBp, float* __restrict__ colA,
                          float* __restrict__ invColA, float* __restrict__ invRowBp) {
  int idx = blockIdx.x * blockDim.x + threadIdx.x;

  // rowBp[nt(4)][ks(64)][lane(32)][j(2)] : fragment order for row_dft waves.
  // Value = B[k=ks*4+2*half+j][n=nt*16+nc] of e^{-i 2pi w kf/256} basis,
  // n=2*kf+c: re=cos, im=-sin.
  if (idx < 16384) {
    int nt = idx >> 12, rem = idx & 4095;
    int ks = rem >> 6, l = (rem >> 1) & 31, j = rem & 1;
    int half = l >> 4, nc = l & 15;
    int w = ks * 4 + 2 * half + j;
    int n = nt * 16 + nc;
    int kf = n >> 1, c = n & 1;
    float th = TWO_PI * (float)((w * kf) & 255) * (1.0f / 256.0f);
    rowBp[idx] = c ? -__sinf(th) : __cosf(th);
  }
  // colA[r][h][c]: kx = r<32 ? r : 192+r (rows 224..255); e^{-i 2pi h kx/256}
  if (idx < 64 * 256 * 2) {
    int r = idx >> 9, rem = idx & 511, h = rem >> 1, c = rem & 1;
    int kx = (r < 32) ? r : (192 + r);
    float th = TWO_PI * (float)((h * kx) & 255) * (1.0f / 256.0f);
    colA[idx] = c ? -__sinf(th) : __cosf(th);
  }
  // invColA[h][r][c]: e^{+i 2pi h kx/256} / 256
  if (idx < 256 * 64 * 2) {
    int h = idx >> 7, rem = idx & 127, r = rem >> 1, c = rem & 1;
    int kx = (r < 32) ? r : (192 + r);
    float th = TWO_PI * (float)((h * kx) & 255) * (1.0f / 256.0f);
    invColA[idx] = (c ? __sinf(th) : __cosf(th)) * (1.0f / 256.0f);
  }
  // invRowBp[nt(16)][ks(16)][lane(32)][j(2)] : fragment order for inv_row.
  // K index kd = ks*4+2*half+j in [0,64) = 2*kf+c; N index w = nt*16+nc.
  // even kd: fac*cos, odd kd: -fac*sin ; fac = (kf==0?1:2)/256
  if (idx < 16384) {
    int nt = idx >> 10, rem = idx & 1023;
    int ks = rem >> 6, l = (rem >> 1) & 31, j = rem & 1;
    int half = l >> 4, nc = l & 15;
    int kd = ks * 4 + 2 * half + j;
    int w = nt * 16 + nc;
    int kf = kd >> 1, c = kd & 1;
    float th = TWO_PI * (float)((w * kf) & 255) * (1.0f / 256.0f);
    float fac = ((kf == 0) ? 1.0f : 2.0f) * (1.0f / 256.0f);
    invRowBp[idx] = c ? (-fac * __sinf(th)) : (fac * __cosf(th));
  }
}

// ---------------------------------------------------------------------------
// Kernel 2: spectral weights = weights(2048x64) @ mlp_w^T(64x2048),
// scattered into point-major layout Wc[s][kx][ky][ci][co][2].
// ---------------------------------------------------------------------------
__global__ void mlp_gemm(const float* __restrict__ Wts, const float* __restrict__ MlpW,
                         float* __restrict__ Wc) {
  int lane = threadIdx.x & 31, wv = threadIdx.x >> 5;
  int half = lane >> 4, mr = lane & 15, nc = lane & 15;
  int m0 = blockIdx.y * 16;
  int n0 = blockIdx.x * 64 + wv * 16;
  v8f acc = {};
#pragma unroll
  for (int k = 0; k < 64; k += 4) {
    int ka = k + 2 * half;
    v2f a, b;
    a.x = Wts[(m0 + mr) * 64 + ka];
    a.y = Wts[(m0 + mr) * 64 + ka + 1];
    b.x = MlpW[(n0 + nc) * 64 + ka];      // B[k][n] = mlp_w[n][k]
    b.y = MlpW[(n0 + nc) * 64 + ka + 1];
    acc = wmma4(a, b, acc);
  }
  // col index -> (kx, ky, comp); row index -> (ci, co, s)
  int ccol = n0 + nc;
  int comp = ccol & 1, ky = (ccol >> 1) & 31, kx = ccol >> 6;
#pragma unroll
  for (int j = 0; j < 8; ++j) {
    int m = m0 + j + 8 * half;
    int s = m & 1, co = (m >> 1) & 31, ci = m >> 6;
    size_t o = ((((size_t)((s * 32 + kx) * 32 + ky)) * 32 + ci) * 32 + co) * 2 + comp;
    Wc[o] = acc[j];
  }
}

// ---------------------------------------------------------------------------
// Kernel 3: truncated rfft along W as GEMM: (65536x256) @ rowB(256x64)
// A tile (16x256 f32) staged into LDS via TDM (fallback: cooperative loads).
// ---------------------------------------------------------------------------
__global__ void row_dft(const float* __restrict__ X, const float* __restrict__ rowBp,
                        float* __restrict__ Xw) {
  __shared__ float As[16 * 256];
  int R0 = blockIdx.x * 16;
#if HAVE_TDM
  if (threadIdx.x < 32) {
    tdm_load_2d_f32(X + (size_t)R0 * 256, (unsigned)(size_t)As, 256, 16);
    __builtin_amdgcn_s_wait_tensorcnt(0);
  }
  __syncthreads();
#else
  for (int i = threadIdx.x; i < 16 * 256; i += blockDim.x)
    As[i] = X[(size_t)R0 * 256 + i];
  __syncthreads();
#endif
  int lane = threadIdx.x & 31, wv = threadIdx.x >> 5;
  int half = lane >> 4, mr = lane & 15, nc = lane & 15;
  const float* Bp = rowBp + (size_t)wv * 4096;  // this wave's N-tile, [64][32][2]
  v8f acc = {};
#pragma unroll 4
  for (int ks = 0; ks < 64; ++ks) {
    v2f a, b;
    a.x = As[mr * 256 + ks * 4 + 2 * half];
    a.y = As[mr * 256 + ks * 4 + 2 * half + 1];
    b.x = Bp[ks * 64 + 2 * lane];
    b.y = Bp[ks * 64 + 2 * lane + 1];
    acc = wmma4(a, b, acc);
  }
#pragma unroll
  for (int j = 0; j < 8; ++j)
    Xw[(size_t)(R0 + j + 8 * half) * 64 + wv * 16 + nc] = acc[j];
}

// ---------------------------------------------------------------------------
// Kernel 4: column DFT (complex GEMM): Xf(64x32) = colA(64x256) @ Xw_bc(256x32)
// One block per (b,ci); 8 waves = 4 M-tiles x 2 N-tiles.
// Output scattered point-major: Xf[(r*32+ky)][b][ci][2].
// ---------------------------------------------------------------------------
__global__ void col_dft(const float* __restrict__ colA, const float* __restrict__ Xw,
                        float* __restrict__ Xf) {
  int bc = blockIdx.x;
  int lane = threadIdx.x & 31, wv = threadIdx.x >> 5;
  int half = lane >> 4, mr = lane & 15, nc = lane & 15;
  int m0 = (wv >> 1) * 16, n0 = (wv & 1) * 16;
  const float* Bx = Xw + (size_t)bc * 256 * 64;  // [h][64] (re,im interleaved)
  v8f accR = {}, accI = {};
#pragma unroll 2
  for (int k = 0; k < 256; k += 4) {
    int ka = k + 2 * half;
    v2f ar, ai, nai, br, bi;
    ar.x = colA[((m0 + mr) * 256 + ka) * 2 + 0];   // 4 contiguous floats -> b128
    ai.x = colA[((m0 + mr) * 256 + ka) * 2 + 1];
    ar.y = colA[((m0 + mr) * 256 + ka) * 2 + 2];
    ai.y = colA[((m0 + mr) * 256 + ka) * 2 + 3];
    br.x = Bx[ka * 64 + (n0 + nc) * 2 + 0];        // b64 pairs
    bi.x = Bx[ka * 64 + (n0 + nc) * 2 + 1];
    br.y = Bx[(ka + 1) * 64 + (n0 + nc) * 2 + 0];
    bi.y = Bx[(ka + 1) * 64 + (n0 + nc) * 2 + 1];
    nai.x = -ai.x; nai.y = -ai.y;
    accR = wmma4(ar, br, accR);
    accR = wmma4(nai, bi, accR);
    accI = wmma4(ar, bi, accI);
    accI = wmma4(ai, br, accI);
  }
  int b = bc >> 5, ci = bc & 31;
#pragma unroll
  for (int j = 0; j < 8; ++j) {
    int r = m0 + j + 8 * half;
    size_t o = ((size_t)(r * 32 + (n0 + nc)) * 8 + b) * 64 + ci * 2;
    Xf[o + 0] = accR[j];
    Xf[o + 1] = accI[j];
  }
}

// ---------------------------------------------------------------------------
// Kernel 5: per-frequency-point modulated channel mix.
// out[b,o] = mod_s[b,pt] * sum_i Xf[pt][b,i] * Wc[pt][i,o]   (complex)
// One block per point (2048); 2 waves (N-tiles); M=16 (B=8 padded).
// A block (8x32 complex) and B block (32x32 complex) are contiguous per point.
// ---------------------------------------------------------------------------
__global__ void spectral_mix(const float* __restrict__ Xf, const float* __restrict__ Wc,
                             const float* __restrict__ mod1, const float* __restrict__ mod2,
                             float* __restrict__ Yf) {
  int pt = blockIdx.x;
  int s = pt >> 10, p = pt & 1023, kx = p >> 5, ky = p & 31;
  int lane = threadIdx.x & 31, wv = threadIdx.x >> 5;
  int half = lane >> 4, mr = lane & 15, nc = lane & 15;
  int n0 = wv * 16;
  int r = s * 32 + kx;  // stored spectral row (32..63 <=> kx 224..255)
  const float* mod = s ? mod2 : mod1;
  const float* Ap = Xf + (size_t)(r * 32 + ky) * 512;               // [8][32][2]
  const float* Wp = Wc + (size_t)((s * 32 + kx) * 32 + ky) * 2048;  // [32][32][2]
  v8f accR = {}, accI = {};
#pragma unroll
  for (int k = 0; k < CI; k += 4) {
    int ka = k + 2 * half;
    float arx = 0.f, aix = 0.f, ary = 0.f, aiy = 0.f;
    if (mr < BATCH) {
      const float* ap = Ap + (mr * 32 + ka) * 2;  // 4 contiguous -> b128
      arx = ap[0]; aix = ap[1]; ary = ap[2]; aiy = ap[3];
    }
    v2f ar, ai, nai, br, bi;
    ar.x = arx; ar.y = ary;
    ai.x = aix; ai.y = aiy;
    nai.x = -aix; nai.y = -aiy;
    br.x = Wp[(ka * 32 + n0 + nc) * 2 + 0];        // b64 pairs
    bi.x = Wp[(ka * 32 + n0 + nc) * 2 + 1];
    br.y = Wp[((ka + 1) * 32 + n0 + nc) * 2 + 0];
    bi.y = Wp[((ka + 1) * 32 + n0 + nc) * 2 + 1];
    accR = wmma4(ar, br, accR);
    accR = wmma4(nai, bi, accR);
    accI = wmma4(ar, bi, accI);
    accI = wmma4(ai, br, accI);
  }
#pragma unroll
  for (int j = 0; j < 8; ++j) {
    int b = j + 8 * half;
    if (b < BATCH) {
      float m_r = mod[((b * 32 + kx) * 32 + ky) * 2 + 0];
      float m_i = mod[((b * 32 + kx) * 32 + ky) * 2 + 1];
      float cr = accR[j], ci = accI[j];
      size_t o = (size_t)(((b * CO + (n0 + nc)) * 64 + r) * 32 + ky) * 2;
      Yf[o + 0] = cr * m_r - ci * m_i;
      Yf[o + 1] = cr * m_i + ci * m_r;
    }
  }
}

// ---------------------------------------------------------------------------
// Kernel 6: inverse column transform (complex GEMM):
// Z(256x32) = invColA(256x64) @ Yf_bo(64x32), per (b,o)
// ---------------------------------------------------------------------------
__global__ void inv_col(const float* __restrict__ invColA, const float* __restrict__ Yf,
                        float* __restrict__ Z) {
  int bo = blockIdx.x;
  int lane = threadIdx.x & 31, wv = threadIdx.x >> 5;
  int half = lane >> 4, mr = lane & 15, nc = lane & 15;
  int m0 = (blockIdx.y * 4 + (wv >> 1)) * 16, n0 = (wv & 1) * 16;
  const float* By = Yf + (size_t)bo * 4096;  // [r(64)][ky(32)][2]
  v8f accR = {}, accI = {};
#pragma unroll
  for (int k = 0; k < 64; k += 4) {
    int ka = k + 2 * half;
    v2f ar, ai, nai, br, bi;
    ar.x = invColA[((m0 + mr) * 64 + ka) * 2 + 0];   // b128
    ai.x = invColA[((m0 + mr) * 64 + ka) * 2 + 1];
    ar.y = invColA[((m0 + mr) * 64 + ka) * 2 + 2];
    ai.y = invColA[((m0 + mr) * 64 + ka) * 2 + 3];
    br.x = By[(ka * 32 + n0 + nc) * 2 + 0];          // b64 pairs
    bi.x = By[(ka * 32 + n0 + nc) * 2 + 1];
    br.y = By[((ka + 1) * 32 + n0 + nc) * 2 + 0];
    bi.y = By[((ka + 1) * 32 + n0 + nc) * 2 + 1];
    nai.x = -ai.x; nai.y = -ai.y;
    accR = wmma4(ar, br, accR);
    accR = wmma4(nai, bi, accR);
    accI = wmma4(ar, bi, accI);
    accI = wmma4(ai, br, accI);
  }
  float* Zp = Z + (size_t)bo * 16384;  // [h(256)][ky(32)][2]
#pragma unroll
  for (int j = 0; j < 8; ++j) {
    int h = m0 + j + 8 * half;
    Zp[(h * 32 + n0 + nc) * 2 + 0] = accR[j];
    Zp[(h * 32 + n0 + nc) * 2 + 1] = accI[j];
  }
}

// ---------------------------------------------------------------------------
// Kernel 7: inverse real row transform: out(65536x256) = Z(65536x64) @ invRowB(64x256)
// A tile (16x64 f32) via TDM into LDS; fragment-packed B.
// ---------------------------------------------------------------------------
__global__ void inv_row(const float* __restrict__ Z, const float* __restrict__ invRowBp,
                        float* __restrict__ Out) {
  __shared__ float As[16 * 64];
  int R0 = blockIdx.x * 16;
#if HAVE_TDM
  if (threadIdx.x < 32) {
    tdm_load_2d_f32(Z + (size_t)R0 * 64, (unsigned)(size_t)As, 64, 16);
    __builtin_amdgcn_s_wait_tensorcnt(0);
  }
  __syncthreads();
#else
  for (int i = threadIdx.x; i < 16 * 64; i += blockDim.x)
    As[i] = Z[(size_t)R0 * 64 + i];
  __syncthreads();
#endif
  int lane = threadIdx.x & 31, wv = threadIdx.x >> 5;
  int half = lane >> 4, mr = lane & 15, nc = lane & 15;
  int nt = blockIdx.y * 4 + wv;
  const float* Bp = invRowBp + (size_t)nt * 1024;  // [16][32][2]
  v8f acc = {};
#pragma unroll
  for (int ks = 0; ks < 16; ++ks) {
    v2f a, b;
    a.x = As[mr * 64 + ks * 4 + 2 * half];
    a.y = As[mr * 64 + ks * 4 + 2 * half + 1];
    b.x = Bp[ks * 64 + 2 * lane];
    b.y = Bp[ks * 64 + 2 * lane + 1];
    acc = wmma4(a, b, acc);
  }
#pragma unroll
  for (int j = 0; j < 8; ++j)
    Out[(size_t)(R0 + j + 8 * half) * 256 + nt * 16 + nc] = acc[j];
}

// ---------------------------------------------------------------------------
extern "C" void kernel_launch(void* const* d_in, const int* in_sizes, int n_in,
                              void* d_out, int out_size, void* d_ws, size_t ws_size,
                              hipStream_t stream) {
  const float* x     = (const float*)d_in[0];  // (8,32,256,256)
  const float* mod1  = (const float*)d_in[1];  // (8,1,1,32,32,2)
  const float* mod2  = (const float*)d_in[2];  // (8,1,1,32,32,2)
  const float* wts   = (const float*)d_in[3];  // (32,32,2,64)
  const float* mlp_w = (const float*)d_in[4];  // (2048,64)
  float* out = (float*)d_out;                  // (8,32,256,256)

  float* ws = (float*)d_ws;
  float* rowBp    = ws + OFF_ROWB;
  float* colA     = ws + OFF_COLA;
  float* invColA  = ws + OFF_ICOLA;
  float* invRowBp = ws + OFF_IROWB;
  float* Wc       = ws + OFF_WC;
  float* Xw       = ws + OFF_XW;
  float* Xf       = ws + OFF_XF;
  float* Yf       = ws + OFF_YF;
  float* Z        = ws + OFF_Z;   // reuses Xw (consumed before Z is written)

  // 1. DFT basis tables (fragment-swizzled)
  gen_basis<<<128, 256, 0, stream>>>(rowBp, colA, invColA, invRowBp);
  // 2. Spectral weights via MLP GEMM -> point-major layout
  mlp_gemm<<<dim3(32, 128), 128, 0, stream>>>(wts, mlp_w, Wc);
  // 3. Truncated rfft along W (TDM-fed LDS tiles)
  row_dft<<<4096, 128, 0, stream>>>(x, rowBp, Xw);
  // 4. Truncated DFT along H (corner rows only) -> point-major spectrum
  col_dft<<<256, 256, 0, stream>>>(colA, Xw, Xf);
  // 5. Modulated channel mix per frequency point
  spectral_mix<<<2048, 64, 0, stream>>>(Xf, Wc, mod1, mod2, Yf);
  // 6. Inverse column transform
  inv_col<<<dim3(256, 4), 256, 0, stream>>>(invColA, Yf, Z);
  // 7. Inverse real row transform -> final output (TDM-fed LDS tiles)
  inv_row<<<dim3(4096, 4), 128, 0, stream>>>(Z, invRowBp, out);
}